// DeltaEncoder_71579924955244
// MI455X (gfx1250) — compile-verified
//
#include <hip/hip_runtime.h>
#include <cstdint>

// Problem constants (from reference): B=32, T=512, F=64, O=64
#define B_ 32
#define T_ 512
#define F_ 64
#define O_ 64

#define TC 32                       // time steps per chunk
#define NCHUNK (T_ / TC)            // 16
#define THREADS 256                 // 8 wave32 waves; 4 o x 64 f chains per block
#define OTILE 4
#define ROWSTRIDE 36                // floats; 144B row stride (16B aligned, spreads banks)

#define LDS_IN_FLOATS (TC * F_)                                   // 2048 floats = 8KB
#define LDS_TOTAL_BYTES ((LDS_IN_FLOATS + THREADS * ROWSTRIDE) * 4)  // 45056 B

extern "C" __global__ __launch_bounds__(THREADS)
void lif_delta_encoder_kernel(const float* __restrict__ in,     // [B,T,F]
                              const float* __restrict__ enc_w,  // [O,1]
                              const float* __restrict__ enc_b,  // [O]
                              const float* __restrict__ bn_w,
                              const float* __restrict__ bn_b,
                              const float* __restrict__ bn_mean,
                              const float* __restrict__ bn_var,
                              float* __restrict__ out)          // [B,O,F,T]
{
    extern __shared__ float smem[];
    float* sIn  = smem;                    // [TC][F_]   raw input rows
    float* sOut = smem + LDS_IN_FLOATS;    // [THREADS][ROWSTRIDE] spike tile

    const int tid = threadIdx.x;
    const int b   = (int)blockIdx.x >> 4;          // 0..31
    const int o0  = ((int)blockIdx.x & 15) * OTILE; // o tile base
    const int o   = o0 + (tid >> 6);               // this thread's output channel
    const int f   = tid & 63;                      // this thread's feature

    // Fold BN (eval) + Linear(1,O) into a per-channel affine: x = delta*A + C
    const float scale = bn_w[0] * rsqrtf(bn_var[0] + 1e-5f);
    const float w     = enc_w[o];
    const float A     = scale * w;
    const float C     = (bn_b[0] - bn_mean[0] * scale) * w + enc_b[o];

    const float* inB = in + (size_t)b * (T_ * F_);

    // Seed prev with the t=0 raw sample -> delta(t=0) == 0 by construction
    float prev = inB[f];
    float v    = 0.0f;

    // LDS byte offsets: low 32 bits of a generic pointer into __shared__ == LDS offset
    const uint32_t ldsInBase  = (uint32_t)(uintptr_t)sIn;
    const uint32_t ldsOutBase = (uint32_t)(uintptr_t)sOut;

    for (int c = 0; c < NCHUNK; ++c) {
        // ---- async stage input chunk: in[b, c*TC .. c*TC+31, :] is 8KB contiguous ----
        {
            uint64_t g = (uint64_t)(uintptr_t)(inB + c * (TC * F_)) + (uint64_t)((uint32_t)tid * 32u);
            uint32_t l = ldsInBase + (uint32_t)tid * 32u;
            asm volatile("global_load_async_to_lds_b128 %0, %1, off"
                         :: "v"(l), "v"(g) : "memory");
            asm volatile("global_load_async_to_lds_b128 %0, %1, off offset:16"
                         :: "v"(l), "v"(g) : "memory");
        }
        asm volatile("s_wait_asynccnt 0" ::: "memory");
        __syncthreads();

        // ---- LIF scan over TC steps for this (b,o,f) chain ----
        #pragma unroll
        for (int i = 0; i < TC; i += 4) {
            float4 s4;
            float* sp = &s4.x;
            #pragma unroll
            for (int j = 0; j < 4; ++j) {
                float raw = sIn[(i + j) * F_ + f];
                float x   = (raw - prev) * A + C;   // BN + Linear on delta
                prev      = raw;
                float h   = 0.5f * (v + x);          // charge: v + (x - v)/TAU, TAU=2
                bool  fire = (h >= 1.0f);
                sp[j]     = fire ? 1.0f : 0.0f;      // Heaviside spike
                v         = fire ? 0.0f : h;         // hard reset
            }
            *(float4*)&sOut[tid * ROWSTRIDE + i] = s4;
        }
        __syncthreads();

        // ---- coalesced async writeout: 256 rows x 128B, 8 lanes per row ----
        #pragma unroll
        for (int p = 0; p < 8; ++p) {
            int idx = p * THREADS + tid;
            int r   = idx >> 3;          // row = (o-o0)*64 + f
            int q   = idx & 7;           // 16B segment within the 128B row
            int oo  = o0 + (r >> 6);
            int ff  = r & 63;
            size_t   go = (((size_t)(b * O_ + oo) * F_ + ff) * T_ + (size_t)c * TC + (size_t)q * 4);
            uint64_t g  = (uint64_t)(uintptr_t)(out + go);
            uint32_t l  = ldsOutBase + (uint32_t)(r * ROWSTRIDE + q * 4) * 4u;
            // Non-temporal: 256MB stream should not thrash the 192MB L2
            asm volatile("global_store_async_from_lds_b128 %0, %1, off th:TH_STORE_NT"
                         :: "v"(g), "v"(l) : "memory");
        }
        asm volatile("s_wait_asynccnt 0" ::: "memory");
        __syncthreads();
    }
}

extern "C" void kernel_launch(void* const* d_in, const int* in_sizes, int n_in,
                              void* d_out, int out_size, void* d_ws, size_t ws_size,
                              hipStream_t stream) {
    (void)in_sizes; (void)n_in; (void)out_size; (void)d_ws; (void)ws_size;
    const float* in      = (const float*)d_in[0];
    const float* enc_w   = (const float*)d_in[1];
    const float* enc_b   = (const float*)d_in[2];
    const float* bn_w    = (const float*)d_in[3];
    const float* bn_b    = (const float*)d_in[4];
    const float* bn_mean = (const float*)d_in[5];
    const float* bn_var  = (const float*)d_in[6];
    float* out = (float*)d_out;

    dim3 grid(B_ * (O_ / OTILE));   // 32 * 16 = 512 blocks
    dim3 block(THREADS);            // 256 threads = 8 wave32 waves
    lif_delta_encoder_kernel<<<grid, block, LDS_TOTAL_BYTES, stream>>>(
        in, enc_w, enc_b, bn_w, bn_b, bn_mean, bn_var, out);
}